// GraphEncoder_60464549593088
// MI455X (gfx1250) — compile-verified
//
#include <hip/hip_runtime.h>
#include <math.h>

typedef float v2f __attribute__((ext_vector_type(2)));
typedef float v8f __attribute__((ext_vector_type(8)));

// ---------------------------------------------------------------------------
// Utility kernels
// ---------------------------------------------------------------------------
__global__ void k_fill0(float* __restrict__ p, size_t n) {
  size_t i = (size_t)blockIdx.x * blockDim.x + threadIdx.x;
  if (i < n) p[i] = 0.0f;
}

__global__ void k_count(const long long* __restrict__ edges,
                        float* __restrict__ cnt, int nE) {
  int e = blockIdx.x * blockDim.x + threadIdx.x;
  if (e < nE) {
    long long dst = edges[2 * (long long)e + 1];
    atomicAdd(&cnt[dst], 1.0f);
  }
}

// wave-per-edge scatter: AGG[dst] += H[src]
__global__ void k_scatter(const float* __restrict__ H,
                          const long long* __restrict__ edges,
                          float* __restrict__ AGG, int nE, int D) {
  int w    = (int)((blockIdx.x * (size_t)blockDim.x + threadIdx.x) >> 5);
  int lane = threadIdx.x & 31;
  if (w >= nE) return;
  long long src = edges[2 * (long long)w + 0];
  long long dst = edges[2 * (long long)w + 1];
  const float* hs = H + (size_t)src * D;
  float*       ad = AGG + (size_t)dst * D;
  for (int j = lane; j < D; j += 32) atomicAdd(&ad[j], hs[j]);
}

// wave-per-row L2 normalize (supports IN == OUT)
__global__ void k_rownorm(const float* __restrict__ IN, float* __restrict__ OUT, int D) {
  int n    = blockIdx.x;
  int lane = threadIdx.x;
  const float* in = IN + (size_t)n * D;
  float s = 0.0f;
  for (int j = lane; j < D; j += 32) { float v = in[j]; s += v * v; }
#pragma unroll
  for (int off = 16; off > 0; off >>= 1) s += __shfl_xor(s, off, 32);
  float inv = 1.0f / fmaxf(sqrtf(s), 1e-12f);
  float* out = OUT + (size_t)n * D;
  for (int j = lane; j < D; j += 32) out[j] = in[j] * inv;
}

// ---------------------------------------------------------------------------
// WMMA fp32 GEMM tiles (one 16x16 D tile per wave32)
//
// Y[n,m] = sum_k X[n,k] * W[m,k]   (X: NxK row-major, W: MxK row-major)
// A frag (16x4 f32): lane l -> row (l&15), K pair base ((l>>4)*2)
// B frag (4x16 f32): lane l -> col (l&15), same K pair base
// D (16x16 f32): vgpr v, lane l -> row v + 8*(l>>4), col (l&15)
// ---------------------------------------------------------------------------
template <int K>
__global__ __launch_bounds__(256)
void k_gemm_relu(const float* __restrict__ X, const float* __restrict__ W,
                 const float* __restrict__ bias, float* __restrict__ Y,
                 int N, int M) {
  int wave = (int)((blockIdx.x * (size_t)blockDim.x + threadIdx.x) >> 5);
  int lane = threadIdx.x & 31;
  int tilesN = M >> 4;
  int total  = (N >> 4) * tilesN;
  if (wave >= total) return;
  int tm = wave / tilesN, tn = wave - tm * tilesN;
  int r   = lane & 15;
  int row = (tm << 4) + r;
  int col = (tn << 4) + r;
  int kh  = (lane >> 4) << 1;
  const float* xr = X + (size_t)row * K + kh;
  const float* wr = W + (size_t)col * K + kh;
  v8f acc = {0.f, 0.f, 0.f, 0.f, 0.f, 0.f, 0.f, 0.f};
#pragma unroll
  for (int k = 0; k < K; k += 4) {
    v2f a = *(const v2f*)(xr + k);
    v2f b = *(const v2f*)(wr + k);
    acc = __builtin_amdgcn_wmma_f32_16x16x4_f32(false, a, false, b,
                                                (short)0, acc, false, false);
  }
  float bv = bias[col];
  int rbase = (tm << 4) + ((lane >> 4) << 3);
#pragma unroll
  for (int v = 0; v < 8; ++v)
    Y[(size_t)(rbase + v) * M + col] = fmaxf(acc[v] + bv, 0.0f);
}

// OUT = (AGG/deg) @ Wl^T + bl + X @ Wr^T   (all K-wide inputs, M-wide output)
template <int K>
__global__ __launch_bounds__(256)
void k_gemm_out(const float* __restrict__ AGG, const float* __restrict__ cnt,
                const float* __restrict__ Wl, const float* __restrict__ bl,
                const float* __restrict__ X, const float* __restrict__ Wr,
                float* __restrict__ OUT, int N, int M) {
  int wave = (int)((blockIdx.x * (size_t)blockDim.x + threadIdx.x) >> 5);
  int lane = threadIdx.x & 31;
  int tilesN = M >> 4;
  int total  = (N >> 4) * tilesN;
  if (wave >= total) return;
  int tm = wave / tilesN, tn = wave - tm * tilesN;
  int r   = lane & 15;
  int row = (tm << 4) + r;
  int col = (tn << 4) + r;
  int kh  = (lane >> 4) << 1;

  float inv = 1.0f / fmaxf(cnt[row], 1.0f);  // fold segment-mean divide into A

  v8f acc = {0.f, 0.f, 0.f, 0.f, 0.f, 0.f, 0.f, 0.f};
  {
    const float* ar = AGG + (size_t)row * K + kh;
    const float* wl = Wl + (size_t)col * K + kh;
#pragma unroll
    for (int k = 0; k < K; k += 4) {
      v2f a = *(const v2f*)(ar + k);
      a.x *= inv; a.y *= inv;
      v2f b = *(const v2f*)(wl + k);
      acc = __builtin_amdgcn_wmma_f32_16x16x4_f32(false, a, false, b,
                                                  (short)0, acc, false, false);
    }
  }
  {
    const float* xr = X + (size_t)row * K + kh;
    const float* wr = Wr + (size_t)col * K + kh;
#pragma unroll
    for (int k = 0; k < K; k += 4) {
      v2f a = *(const v2f*)(xr + k);
      v2f b = *(const v2f*)(wr + k);
      acc = __builtin_amdgcn_wmma_f32_16x16x4_f32(false, a, false, b,
                                                  (short)0, acc, false, false);
    }
  }
  float bv = bl[col];
  int rbase = (tm << 4) + ((lane >> 4) << 3);
#pragma unroll
  for (int v = 0; v < 8; ++v)
    OUT[(size_t)(rbase + v) * M + col] = acc[v] + bv;
}

// ---------------------------------------------------------------------------
// Host-side K dispatch
// ---------------------------------------------------------------------------
static void launch_gemm_relu(int K, int blocks, hipStream_t s,
                             const float* X, const float* W, const float* b,
                             float* Y, int N, int M) {
  switch (K) {
    case 64:  k_gemm_relu<64><<<blocks, 256, 0, s>>>(X, W, b, Y, N, M); break;
    case 128: k_gemm_relu<128><<<blocks, 256, 0, s>>>(X, W, b, Y, N, M); break;
    case 256: k_gemm_relu<256><<<blocks, 256, 0, s>>>(X, W, b, Y, N, M); break;
  }
}

static void launch_gemm_out(int K, int blocks, hipStream_t s,
                            const float* AGG, const float* cnt,
                            const float* Wl, const float* bl,
                            const float* X, const float* Wr,
                            float* OUT, int N, int M) {
  switch (K) {
    case 64:  k_gemm_out<64><<<blocks, 256, 0, s>>>(AGG, cnt, Wl, bl, X, Wr, OUT, N, M); break;
    case 128: k_gemm_out<128><<<blocks, 256, 0, s>>>(AGG, cnt, Wl, bl, X, Wr, OUT, N, M); break;
    case 256: k_gemm_out<256><<<blocks, 256, 0, s>>>(AGG, cnt, Wl, bl, X, Wr, OUT, N, M); break;
  }
}

extern "C" void kernel_launch(void* const* d_in, const int* in_sizes, int n_in,
                              void* d_out, int out_size, void* d_ws, size_t ws_size,
                              hipStream_t stream) {
  const int N  = in_sizes[0] / 128;   // 50000 nodes
  const int nE = in_sizes[1] / 2;     // 300000 edges
  const float* x = (const float*)d_in[0];
  const long long* edges = (const long long*)d_in[1];

  // workspace layout (floats): cnt[N] | H[N*256] | AGG[N*256] | XA[N*256] | XB[N*256]
  float* cnt = (float*)d_ws;
  float* H   = cnt + N;
  float* AGG = H + (size_t)N * 256;
  float* XA  = AGG + (size_t)N * 256;
  float* XB  = XA + (size_t)N * 256;

  // in-degree counts (same for every layer)
  k_fill0<<<(N + 255) / 256, 256, 0, stream>>>(cnt, (size_t)N);
  k_count<<<(nE + 255) / 256, 256, 0, stream>>>(edges, cnt, nE);

  const int dims[6] = {128, 64, 128, 256, 256, 576};  // chained din/dout
  const float* Xin = x;
  for (int L = 0; L < 5; ++L) {
    const int K = dims[L];
    const int M = dims[L + 1];
    const float* Wp = (const float*)d_in[2 + L * 5 + 0];
    const float* bp = (const float*)d_in[2 + L * 5 + 1];
    const float* Wl = (const float*)d_in[2 + L * 5 + 2];
    const float* bl = (const float*)d_in[2 + L * 5 + 3];
    const float* Wr = (const float*)d_in[2 + L * 5 + 4];
    float* OUT = (L == 4) ? (float*)d_out : H;                    // reuse H post-scatter
    float* XN  = (L == 4) ? (float*)d_out : ((L & 1) ? XB : XA);  // ping-pong

    // 1) H = relu(Xin @ Wp^T + bp)    [N x K]
    {
      int tiles = (N / 16) * (K / 16);
      launch_gemm_relu(K, (tiles + 7) / 8, stream, Xin, Wp, bp, H, N, K);
    }
    // 2) AGG = 0                      [N x K]
    {
      size_t ne = (size_t)N * K;
      k_fill0<<<(int)((ne + 255) / 256), 256, 0, stream>>>(AGG, ne);
    }
    // 3) AGG[dst] += H[src] over edges
    k_scatter<<<(nE + 7) / 8, 256, 0, stream>>>(H, edges, AGG, nE, K);
    // 4) OUT = (AGG/deg) @ Wl^T + bl + Xin @ Wr^T   [N x M]
    {
      int tiles = (N / 16) * (M / 16);
      launch_gemm_out(K, (tiles + 7) / 8, stream, AGG, cnt, Wl, bl, Xin, Wr, OUT, N, M);
    }
    // 5) row-wise L2 normalize (in place for final layer into d_out)
    k_rownorm<<<N, 32, 0, stream>>>(OUT, XN, M);

    Xin = XN;
  }
}